// ProposalModule_8100308320901
// MI455X (gfx1250) — compile-verified
//
#include <hip/hip_runtime.h>

#define K_PARTS 24
#define B_SZ    64
#define N_PTS   16384
#define C_CH    128
#define CT      16                 // channels per block tile
#define ROW     (3 + 2 * C_CH)     // 259 output row length
#define SMSTR   17                 // padded stride for the max table

typedef float v2f __attribute__((ext_vector_type(2)));
typedef float v8f __attribute__((ext_vector_type(8)));

// ---------------------------------------------------------------------------
// Kernel 1: per-batch part counts + xyz sums -> stats[b][k][4] = {sx,sy,sz,cnt}
// ---------------------------------------------------------------------------
__global__ __launch_bounds__(256)
void part_stats_kernel(const float* __restrict__ xyz,
                       const int*   __restrict__ labels,
                       float*       __restrict__ stats) {
    __shared__ float s[K_PARTS * 4];
    const int b   = blockIdx.x;
    const int tid = threadIdx.x;
    if (tid < K_PARTS * 4) s[tid] = 0.0f;
    __syncthreads();

    const float* xb = xyz    + (size_t)b * N_PTS * 3;
    const int*   lb = labels + (size_t)b * N_PTS;
    for (int n = tid; n < N_PTS; n += 256) {
        const int   l = lb[n];
        const float x = xb[3 * n + 0];
        const float y = xb[3 * n + 1];
        const float z = xb[3 * n + 2];
        __hip_atomic_fetch_add(&s[4 * l + 0], x,   __ATOMIC_RELAXED, __HIP_MEMORY_SCOPE_WORKGROUP);
        __hip_atomic_fetch_add(&s[4 * l + 1], y,   __ATOMIC_RELAXED, __HIP_MEMORY_SCOPE_WORKGROUP);
        __hip_atomic_fetch_add(&s[4 * l + 2], z,   __ATOMIC_RELAXED, __HIP_MEMORY_SCOPE_WORKGROUP);
        __hip_atomic_fetch_add(&s[4 * l + 3], 1.f, __ATOMIC_RELAXED, __HIP_MEMORY_SCOPE_WORKGROUP);
    }
    __syncthreads();
    if (tid < K_PARTS * 4) stats[b * K_PARTS * 4 + tid] = s[tid];
}

// ---------------------------------------------------------------------------
// Kernel 2: one block per (batch b, 16-channel tile ct).
// Single pass over features: WMMA f32 16x16x4 computes onehot^T x feat
// (segmented sums) on the matrix pipe while ds_max_num_f32 atomics track the
// segmented max. Finalize writes mean_feat, surface (and mean_xyz for ct==0).
// ---------------------------------------------------------------------------
__global__ __launch_bounds__(256)
void seg_wmma_kernel(const float* __restrict__ features,
                     const int*   __restrict__ labels,
                     const float* __restrict__ stats,
                     float*       __restrict__ out) {
    const int ct   = blockIdx.x;          // channel tile 0..7
    const int b    = blockIdx.y;          // batch
    const int tid  = threadIdx.x;
    const int lane = tid & 31;
    const int wv   = tid >> 5;            // wave 0..7, each owns N_PTS/8 points
    const int c    = lane & 15;           // channel-in-tile == part row (low half)
    const int h    = lane >> 4;           // K-pair selector per 32-bit WMMA layout

    __shared__ float ssum[32 * CT];       // [part 0..31][c] segmented sums
    __shared__ float smax[32 * SMSTR];    // [part][c] segmented max, padded

    for (int i = tid; i < 32 * CT;    i += 256) ssum[i] = 0.0f;
    for (int i = tid; i < 32 * SMSTR; i += 256) smax[i] = -__builtin_inff();
    __syncthreads();

    const int    c0   = ct * CT;
    const float* fptr = features + ((size_t)b * C_CH + (c0 + c)) * N_PTS;
    const int*   lptr = labels   + (size_t)b * N_PTS;

    v8f acc0 = {0.f, 0.f, 0.f, 0.f, 0.f, 0.f, 0.f, 0.f};  // parts 0..15
    v8f acc1 = {0.f, 0.f, 0.f, 0.f, 0.f, 0.f, 0.f, 0.f};  // parts 16..31

    const int nbase = wv * (N_PTS / 8);
    #pragma unroll 4
    for (int np = nbase; np < nbase + N_PTS / 8; np += 4) {
        const int idx = np + 2 * h;
        // B fragment (4x16 f32): lane holds B[2h,c],B[2h+1,c] = feat at points
        // idx, idx+1 for channel c -> one 8-byte load.
        const v2f  f   = *(const v2f*)(fptr + idx);
        const int2 lab = *(const int2*)(lptr + idx);

        // A fragments (16x4 f32 onehot): lane holds A[m, 2h] / A[m, 2h+1]
        v2f a0, a1;
        a0.x = (lab.x == c)      ? 1.0f : 0.0f;
        a0.y = (lab.y == c)      ? 1.0f : 0.0f;
        a1.x = (lab.x == c + 16) ? 1.0f : 0.0f;
        a1.y = (lab.y == c + 16) ? 1.0f : 0.0f;

        acc0 = __builtin_amdgcn_wmma_f32_16x16x4_f32(false, a0, false, f,
                                                     (short)0, acc0, false, false);
        acc1 = __builtin_amdgcn_wmma_f32_16x16x4_f32(false, a1, false, f,
                                                     (short)0, acc1, false, false);

        // Segmented running max: lanes 0..15 share a point (same label) and
        // touch consecutive channels -> consecutive LDS banks, conflict-free.
        __hip_atomic_fetch_max(&smax[lab.x * SMSTR + c], f.x,
                               __ATOMIC_RELAXED, __HIP_MEMORY_SCOPE_WORKGROUP);
        __hip_atomic_fetch_max(&smax[lab.y * SMSTR + c], f.y,
                               __ATOMIC_RELAXED, __HIP_MEMORY_SCOPE_WORKGROUP);
    }

    // Fold WMMA accumulators across waves.
    // C/D layout: VGPR r -> M = r + 8*(lane>=16), N = lane&15.
    {
        const int prow = 8 * h;
        #pragma unroll
        for (int r = 0; r < 8; ++r) {
            __hip_atomic_fetch_add(&ssum[(prow + r) * CT + c], acc0[r],
                                   __ATOMIC_RELAXED, __HIP_MEMORY_SCOPE_WORKGROUP);
            const int p2 = prow + r + 16;
            if (p2 < K_PARTS) {      // parts 24..31 are structurally zero
                __hip_atomic_fetch_add(&ssum[p2 * CT + c], acc1[r],
                                       __ATOMIC_RELAXED, __HIP_MEMORY_SCOPE_WORKGROUP);
            }
        }
    }
    __syncthreads();

    // Finalize: mean = sum/denom, surface = max(0, segmax - mean).
    const float* st = stats + (size_t)b * K_PARTS * 4;
    float*       ob = out   + (size_t)b * K_PARTS * ROW;
    for (int t = tid; t < K_PARTS * CT; t += 256) {
        const int   p     = t / CT;
        const int   cc    = t % CT;
        const float cnt   = st[p * 4 + 3];
        const float denom = fmaxf(cnt, 1.0f);
        const float mean  = ssum[p * CT + cc] / denom;
        const float mx    = smax[p * SMSTR + cc];
        ob[p * ROW + 3 + c0 + cc]        = mean;                      // mean_feat
        ob[p * ROW + 3 + C_CH + c0 + cc] = fmaxf(0.0f, mx - mean);    // surface
    }
    if (ct == 0) {
        for (int t = tid; t < K_PARTS * 3; t += 256) {
            const int   p     = t / 3;
            const int   d     = t % 3;
            const float denom = fmaxf(st[p * 4 + 3], 1.0f);
            ob[p * ROW + d] = st[p * 4 + d] / denom;                  // mean_xyz
        }
    }
}

// ---------------------------------------------------------------------------
extern "C" void kernel_launch(void* const* d_in, const int* in_sizes, int n_in,
                              void* d_out, int out_size, void* d_ws, size_t ws_size,
                              hipStream_t stream) {
    const float* xyz      = (const float*)d_in[0];   // [B,N,3]
    const float* features = (const float*)d_in[1];   // [B,C,N]
    const int*   labels   = (const int*)d_in[2];     // [B,N]
    float*       out      = (float*)d_out;           // [B,K,3+2C]
    float*       stats    = (float*)d_ws;            // [B,K,4] scratch

    part_stats_kernel<<<B_SZ, 256, 0, stream>>>(xyz, labels, stats);

    dim3 grid(C_CH / CT, B_SZ);   // (8, 64)
    seg_wmma_kernel<<<grid, 256, 0, stream>>>(features, labels, stats, out);
}